// SelfAttention_73194832658771
// MI455X (gfx1250) — compile-verified
//
#include <hip/hip_runtime.h>
#include <math.h>

// ---- CDNA5 f32 WMMA: D(16x16) = A(16x4) * B(4x16) + C, exact f32 ----
typedef float v2f __attribute__((ext_vector_type(2)));
typedef float v8f __attribute__((ext_vector_type(8)));

#define WMMA_F32(a, b, c) \
  __builtin_amdgcn_wmma_f32_16x16x4_f32(false, (a), false, (b), (short)0, (c), false, false)

static constexpr int NB = 4;      // batch
static constexpr int SQ = 4096;   // sequence
static constexpr int DM = 256;    // model dim
static constexpr int BS = NB * SQ;

// ---------------------------------------------------------------------------
// Kernel 1: q/k/v = in[:, sel*256 : (sel+1)*256] @ W_sel + b_sel
// One wave per 16x16 output tile. grid = (16 col tiles, 1024 row tiles, 3)
// ---------------------------------------------------------------------------
__global__ __launch_bounds__(32) void qkv_kernel(
    const float* __restrict__ in,
    const float* __restrict__ wq, const float* __restrict__ bq,
    const float* __restrict__ wk, const float* __restrict__ bk,
    const float* __restrict__ wv, const float* __restrict__ bv,
    float* __restrict__ qo, float* __restrict__ ko, float* __restrict__ vo)
{
  const int lane = threadIdx.x;
  const int li = lane & 15;          // N / M index within tile
  const int h  = lane >> 4;          // lane half -> K pair select
  const int nt = blockIdx.x;         // feature col tile
  const int mt = blockIdx.y;         // row tile over B*S
  const int sel = blockIdx.z;        // 0=q 1=k 2=v (uniform per block)

  const float* w   = sel == 0 ? wq : (sel == 1 ? wk : wv);
  const float* bia = sel == 0 ? bq : (sel == 1 ? bk : bv);
  float*       out = sel == 0 ? qo : (sel == 1 ? ko : vo);

  const int col = nt * 16 + li;
  // A fragment source: row = mt*16+li of in (row stride 3*DM), K offset 2*h
  const float* arow = in + (size_t)(mt * 16 + li) * (3 * DM) + sel * DM + 2 * h;
  // B fragment source: W[k][col], K offset 2*h
  const float* wcol = w + col + (size_t)(2 * h) * DM;

  v8f c = {0.f, 0.f, 0.f, 0.f, 0.f, 0.f, 0.f, 0.f};
#pragma unroll 8
  for (int kb = 0; kb < DM; kb += 4) {
    v2f a = *(const v2f*)(arow + kb);
    v2f b;
    b.x = wcol[(size_t)kb * DM];
    b.y = wcol[(size_t)(kb + 1) * DM];
    c = WMMA_F32(a, b, c);
  }
  const float bb = bia[col];
  float* op = out + (size_t)(mt * 16) * DM + col;
#pragma unroll
  for (int r = 0; r < 8; ++r)
    op[(size_t)(r + 8 * h) * DM] = c[r] + bb;
}

// ---------------------------------------------------------------------------
// Kernel 2: fused flash attention + output projection.
// One wave per 16-query row block: loops 256 key tiles.
//  S^T = K_tile @ Q^T  (lane owns one query row -> cheap row softmax)
//  online softmax, O += P @ V, finally out = (O/l) @ wo + bo.
// grid = 1024 blocks of 32 threads.
// ---------------------------------------------------------------------------
__global__ __launch_bounds__(32) void attn_kernel(
    const float* __restrict__ qw, const float* __restrict__ kw,
    const float* __restrict__ vw,
    const float* __restrict__ wo, const float* __restrict__ bo,
    float* __restrict__ out)
{
  __shared__ float Pl[16][18];    // P[query][key]  (pad 18: 8B-aligned, few conflicts)
  __shared__ float Ol[16][260];   // O[query][feat] (pad 260: banks offset 4/row)

  const int lane = threadIdx.x;
  const int li = lane & 15;
  const int h  = lane >> 4;
  const int g  = blockIdx.x;      // 0..1023
  const int b  = g >> 8;          // batch
  const int mt = g & 255;         // query row tile within batch

  const size_t rowbase = (size_t)b * SQ + mt * 16;
  // Q used as B-fragment (B[f][n] = Q[n][f]); contiguous float2 per K-step.
  const float* qrow = qw + (rowbase + li) * DM + 2 * h;

  v8f o[16];
#pragma unroll
  for (int f = 0; f < 16; ++f) o[f] = (v8f){0.f,0.f,0.f,0.f,0.f,0.f,0.f,0.f};

  float m_run = -__builtin_inff();
  float l_run = 0.f;
  const float sc = 0.0625f;       // 1/sqrt(256)

  for (int t = 0; t < SQ / 16; ++t) {
    // ---- S^T tile: A = K_tile (M=key, K=feat), B = Q^T ----
    const float* krow = kw + ((size_t)b * SQ + t * 16 + li) * DM + 2 * h;
    v8f s = {0.f,0.f,0.f,0.f,0.f,0.f,0.f,0.f};
#pragma unroll 8
    for (int kb = 0; kb < DM; kb += 4) {
      v2f a  = *(const v2f*)(krow + kb);
      v2f qb = *(const v2f*)(qrow + kb);
      s = WMMA_F32(a, qb, s);
    }
    // lane holds scores for query li vs keys t*16 + 8*h + r
    float p[8];
    float mloc = -__builtin_inff();
#pragma unroll
    for (int r = 0; r < 8; ++r) { p[r] = s[r] * sc; mloc = fmaxf(mloc, p[r]); }
    mloc = fmaxf(mloc, __shfl_xor(mloc, 16));
    const float m_new = fmaxf(m_run, mloc);
    float sum = 0.f;
#pragma unroll
    for (int r = 0; r < 8; ++r) { p[r] = __expf(p[r] - m_new); sum += p[r]; }
    sum += __shfl_xor(sum, 16);
    const float alpha = __expf(m_run - m_new);
    l_run = l_run * alpha + sum;
    m_run = m_new;

    // rescale O accumulators: row of o[f][r] is query (r + 8*h)
#pragma unroll
    for (int r = 0; r < 8; ++r) {
      const float ar = __shfl(alpha, r + 8 * h);
#pragma unroll
      for (int f = 0; f < 16; ++f) o[f][r] *= ar;
    }

    // stage P^T -> Pl[query][key] for A-fragment reads
#pragma unroll
    for (int r = 0; r < 8; ++r) Pl[li][8 * h + r] = p[r];
    __syncthreads();

    // ---- O += P @ V_tile : A = P (M=query, K=key), B = V (key x feat) ----
    v2f aP[4];
#pragma unroll
    for (int ks = 0; ks < 4; ++ks)
      aP[ks] = *(const v2f*)&Pl[li][ks * 4 + 2 * h];

    const float* vbase = vw + ((size_t)b * SQ + t * 16) * DM;
#pragma unroll 4
    for (int f = 0; f < 16; ++f) {
      const float* vcol = vbase + f * 16 + li;
#pragma unroll
      for (int ks = 0; ks < 4; ++ks) {
        const int kk = ks * 4 + 2 * h;
        v2f bf;
        bf.x = vcol[(size_t)kk * DM];
        bf.y = vcol[(size_t)(kk + 1) * DM];
        o[f] = WMMA_F32(aP[ks], bf, o[f]);
      }
    }
    __syncthreads();   // Pl reused next iteration
  }

  // ---- epilogue: out = (O / l_run) @ wo + bo ----
  const float inv = 1.f / l_run;   // lane's query row == li
#pragma unroll
  for (int f = 0; f < 16; ++f)
#pragma unroll
    for (int r = 0; r < 8; ++r)
      Ol[r + 8 * h][f * 16 + li] = o[f][r];
  __syncthreads();

  const float* prow = &Ol[li][2 * h];
  for (int ot = 0; ot < 16; ++ot) {
    const float* wcol = wo + ot * 16 + li + (size_t)(2 * h) * DM;
    v8f c = {0.f,0.f,0.f,0.f,0.f,0.f,0.f,0.f};
#pragma unroll 8
    for (int kb = 0; kb < DM; kb += 4) {
      v2f a = *(const v2f*)(prow + kb);
      a.x *= inv;
      a.y *= inv;
      v2f bf;
      bf.x = wcol[(size_t)kb * DM];
      bf.y = wcol[(size_t)(kb + 1) * DM];
      c = WMMA_F32(a, bf, c);
    }
    const float bb = bo[ot * 16 + li];
    float* op = out + rowbase * DM + ot * 16 + li;
#pragma unroll
    for (int r = 0; r < 8; ++r)
      op[(size_t)(r + 8 * h) * DM] = c[r] + bb;
  }
}

// ---------------------------------------------------------------------------
extern "C" void kernel_launch(void* const* d_in, const int* in_sizes, int n_in,
                              void* d_out, int out_size, void* d_ws, size_t ws_size,
                              hipStream_t stream) {
  (void)in_sizes; (void)n_in; (void)out_size; (void)ws_size;
  const float* in = (const float*)d_in[0];
  const float* wq = (const float*)d_in[1];
  const float* bq = (const float*)d_in[2];
  const float* wk = (const float*)d_in[3];
  const float* bk = (const float*)d_in[4];
  const float* wv = (const float*)d_in[5];
  const float* bv = (const float*)d_in[6];
  const float* wo = (const float*)d_in[7];
  const float* bo = (const float*)d_in[8];

  const size_t mat = (size_t)BS * DM;     // 16384*256 floats = 16 MB
  float* q = (float*)d_ws;                // 48 MB total scratch
  float* k = q + mat;
  float* v = k + mat;

  qkv_kernel<<<dim3(16, BS / 16, 3), 32, 0, stream>>>(
      in, wq, bq, wk, bk, wv, bv, q, k, v);
  attn_kernel<<<dim3(BS / 16), 32, 0, stream>>>(
      q, k, v, wo, bo, (float*)d_out);
}